// ExerciseLSTM_36532991820434
// MI455X (gfx1250) — compile-verified
//
#include <hip/hip_runtime.h>
#include <hip/hip_bf16.h>

typedef __attribute__((ext_vector_type(2))) float v2f;
typedef __attribute__((ext_vector_type(8))) float v8f;

#define B_SZ 2048
#define T_SZ 512
#define IN_SZ 2
#define H_SZ 64
#define H_P 66     // padded LDS row stride for h-state (bank-conflict-free A loads)
#define G_SZ 256   // 4*H
#define NC_SZ 9
#define BT 16      // batch rows per block
#define NTHREADS 512

// Fast activations: single v_exp_f32 + single v_rcp_f32 (no IEEE div expansion).
__device__ __forceinline__ float fast_sigmoid(float x) {
    return __builtin_amdgcn_rcpf(1.0f + __expf(-x));
}
__device__ __forceinline__ float fast_tanh(float x) {
    return 2.0f * __builtin_amdgcn_rcpf(1.0f + __expf(-2.0f * x)) - 1.0f;
}

__launch_bounds__(NTHREADS, 1)
__global__ void lstm2_fused_kernel(const float* __restrict__ x,      // [B,T,2]
                                   const float* __restrict__ w_ih0,  // [256,2]
                                   const float* __restrict__ w_hh0,  // [256,64]
                                   const float* __restrict__ b_ih0,  // [256]
                                   const float* __restrict__ b_hh0,  // [256]
                                   const float* __restrict__ w_ih1,  // [256,64]
                                   const float* __restrict__ w_hh1,  // [256,64]
                                   const float* __restrict__ b_ih1,  // [256]
                                   const float* __restrict__ b_hh1,  // [256]
                                   const float* __restrict__ w_fc,   // [9,64]
                                   const float* __restrict__ b_fc,   // [9]
                                   float* __restrict__ out)          // [B,9]
{
    __shared__ float h1[BT * H_P];        // padded: row stride 66 floats
    __shared__ float h2[BT * H_P];        // padded
    __shared__ float c1[BT * H_SZ];
    __shared__ float c2[BT * H_SZ];
    __shared__ float gates[BT * G_SZ];
    __shared__ float xt[BT * IN_SZ];

    const int tid   = threadIdx.x;
    const int wid   = tid >> 5;        // wave id 0..15
    const int lane  = tid & 31;
    const int half  = lane >> 4;       // 0 or 1 (K-half of the WMMA fragment)
    const int lan16 = lane & 15;
    const int n     = wid * 16 + lan16; // gate column this lane owns (B/C column)
    const int b0    = blockIdx.x * BT;  // first batch row of this block

    // ---- Preload per-wave weight slices (B operands) into registers ----
    // For WMMA f32 16x16x4: B fragment per lane = W[n][4k+2*half .. 4k+2*half+1]
    v2f whh0[16], wih1[16], whh1[16];
#pragma unroll
    for (int k = 0; k < 16; ++k) {
        const int c = 4 * k + 2 * half;
        whh0[k] = *(const v2f*)(w_hh0 + n * H_SZ + c);
        wih1[k] = *(const v2f*)(w_ih1 + n * H_SZ + c);
        whh1[k] = *(const v2f*)(w_hh1 + n * H_SZ + c);
    }
    const float bias0 = b_ih0[n] + b_hh0[n];
    const float bias1 = b_ih1[n] + b_hh1[n];
    const float wi0a  = w_ih0[n * IN_SZ + 0];
    const float wi0b  = w_ih0[n * IN_SZ + 1];

    // ---- Zero-init recurrent state + stage x[:,0,:] ----
    for (int i = tid; i < BT * H_P; i += NTHREADS) { h1[i] = 0.0f; h2[i] = 0.0f; }
    for (int i = tid; i < BT * H_SZ; i += NTHREADS) { c1[i] = 0.0f; c2[i] = 0.0f; }
    if (tid < BT * IN_SZ) {
        const int bb = tid >> 1;
        const int ii = tid & 1;
        xt[tid] = x[(size_t)(b0 + bb) * (T_SZ * IN_SZ) + ii];
    }
    __syncthreads();

    for (int t = 0; t < T_SZ; ++t) {
        // ================= Layer 0: gates = x@Wih0^T + h1@Whh0^T + b =================
        // Two independent WMMA accumulation chains (even/odd k) to halve latency.
        v8f accA, accB;
#pragma unroll
        for (int r = 0; r < 8; ++r) {
            const int m = r + 8 * half;  // C-fragment row for VGPR r
            accA[r] = bias0 + xt[m * IN_SZ + 0] * wi0a + xt[m * IN_SZ + 1] * wi0b;
            accB[r] = 0.0f;
        }
#pragma unroll
        for (int k = 0; k < 8; ++k) {
            const int cA = 4 * (2 * k)     + 2 * half;
            const int cB = 4 * (2 * k + 1) + 2 * half;
            v2f aA = *(const v2f*)(h1 + lan16 * H_P + cA);
            v2f aB = *(const v2f*)(h1 + lan16 * H_P + cB);
            accA = __builtin_amdgcn_wmma_f32_16x16x4_f32(false, aA, false, whh0[2 * k],
                                                         (short)0, accA, false, false);
            accB = __builtin_amdgcn_wmma_f32_16x16x4_f32(false, aB, false, whh0[2 * k + 1],
                                                         (short)0, accB, false, false);
        }
#pragma unroll
        for (int r = 0; r < 8; ++r) {
            const int m = r + 8 * half;
            gates[m * G_SZ + n] = accA[r] + accB[r];
        }
        __syncthreads();

        // elementwise LSTM cell update (layer 0): 1024 elems over 512 threads
#pragma unroll
        for (int j = 0; j < (BT * H_SZ) / NTHREADS; ++j) {
            const int e  = tid + j * NTHREADS;
            const int bb = e >> 6;
            const int hh = e & 63;
            float gi = fast_sigmoid(gates[bb * G_SZ + hh]);
            float gf = fast_sigmoid(gates[bb * G_SZ + 64 + hh]);
            float gg = fast_tanh  (gates[bb * G_SZ + 128 + hh]);
            float go = fast_sigmoid(gates[bb * G_SZ + 192 + hh]);
            float cc = gf * c1[e] + gi * gg;
            c1[e] = cc;
            h1[bb * H_P + hh] = go * fast_tanh(cc);
        }
        __syncthreads();

        // ================= Layer 1: gates = h1@Wih1^T + h2@Whh1^T + b ================
        // Two independent chains: one per source matmul.
#pragma unroll
        for (int r = 0; r < 8; ++r) { accA[r] = bias1; accB[r] = 0.0f; }
#pragma unroll
        for (int k = 0; k < 16; ++k) {
            const int c = 4 * k + 2 * half;
            v2f aA = *(const v2f*)(h1 + lan16 * H_P + c);
            v2f aB = *(const v2f*)(h2 + lan16 * H_P + c);
            accA = __builtin_amdgcn_wmma_f32_16x16x4_f32(false, aA, false, wih1[k],
                                                         (short)0, accA, false, false);
            accB = __builtin_amdgcn_wmma_f32_16x16x4_f32(false, aB, false, whh1[k],
                                                         (short)0, accB, false, false);
        }
#pragma unroll
        for (int r = 0; r < 8; ++r) {
            const int m = r + 8 * half;
            gates[m * G_SZ + n] = accA[r] + accB[r];
        }
        __syncthreads();

        // elementwise LSTM cell update (layer 1) + stage x[:,t+1,:] for next step
#pragma unroll
        for (int j = 0; j < (BT * H_SZ) / NTHREADS; ++j) {
            const int e  = tid + j * NTHREADS;
            const int bb = e >> 6;
            const int hh = e & 63;
            float gi = fast_sigmoid(gates[bb * G_SZ + hh]);
            float gf = fast_sigmoid(gates[bb * G_SZ + 64 + hh]);
            float gg = fast_tanh  (gates[bb * G_SZ + 128 + hh]);
            float go = fast_sigmoid(gates[bb * G_SZ + 192 + hh]);
            float cc = gf * c2[e] + gi * gg;
            c2[e] = cc;
            h2[bb * H_P + hh] = go * fast_tanh(cc);
        }
        if ((t + 1) < T_SZ && tid < BT * IN_SZ) {
            const int bb = tid >> 1;
            const int ii = tid & 1;
            xt[tid] = x[(size_t)(b0 + bb) * (T_SZ * IN_SZ) + (t + 1) * IN_SZ + ii];
        }
        __syncthreads();
    }

    // ================= Final FC on last h2: [16,64]@[64,9]^T + b ================
    if (tid < BT * NC_SZ) {
        const int bb = tid / NC_SZ;
        const int nc = tid % NC_SZ;
        float s = b_fc[nc];
#pragma unroll
        for (int hh = 0; hh < H_SZ; ++hh) {
            s += h2[bb * H_P + hh] * w_fc[nc * H_SZ + hh];
        }
        out[(size_t)(b0 + bb) * NC_SZ + nc] = s;
    }
}

extern "C" void kernel_launch(void* const* d_in, const int* in_sizes, int n_in,
                              void* d_out, int out_size, void* d_ws, size_t ws_size,
                              hipStream_t stream) {
    (void)in_sizes; (void)n_in; (void)out_size; (void)d_ws; (void)ws_size;
    const float* x     = (const float*)d_in[0];
    const float* w_ih0 = (const float*)d_in[1];
    const float* w_hh0 = (const float*)d_in[2];
    const float* b_ih0 = (const float*)d_in[3];
    const float* b_hh0 = (const float*)d_in[4];
    const float* w_ih1 = (const float*)d_in[5];
    const float* w_hh1 = (const float*)d_in[6];
    const float* b_ih1 = (const float*)d_in[7];
    const float* b_hh1 = (const float*)d_in[8];
    const float* w_fc  = (const float*)d_in[9];
    const float* b_fc  = (const float*)d_in[10];
    float* out = (float*)d_out;

    dim3 grid(B_SZ / BT);   // 128 blocks
    dim3 block(NTHREADS);   // 16 wave32s
    lstm2_fused_kernel<<<grid, block, 0, stream>>>(x, w_ih0, w_hh0, b_ih0, b_hh0,
                                                   w_ih1, w_hh1, b_ih1, b_hh1,
                                                   w_fc, b_fc, out);
}